// MultiHeaded_Token_Attention_44727789420585
// MI455X (gfx1250) — compile-verified
//
#include <hip/hip_runtime.h>

// Problem constants from the reference
#define B_ 2
#define L_ 64
#define T_ 32
#define D_ 256

typedef __attribute__((ext_vector_type(2))) float v2f;
typedef __attribute__((ext_vector_type(4))) float v4f;
typedef __attribute__((ext_vector_type(8))) float v8f;

// ---------------------------------------------------------------------------
// Kernel 1: qk[b,i,d] = sum_e Q[b,i,e] * Wk[e,d],  Q = (query·Wq^T + bq)/16
// The Q·bk softmax-shift term is dropped (softmax shift invariance).
// 128 blocks (one per (b,i)), 256 threads.
// ---------------------------------------------------------------------------
__global__ __launch_bounds__(256) void qk_fuse_kernel(
    const float* __restrict__ query, const float* __restrict__ Wq,
    const float* __restrict__ bq, const float* __restrict__ Wk,
    float* __restrict__ qk) {
  __shared__ float Qrow[D_];
  const int m = blockIdx.x;     // b*L + i
  const int e = threadIdx.x;    // 0..255
  const v4f* q4 = (const v4f*)(query + (size_t)m * D_);
  const v4f* w4 = (const v4f*)(Wq + (size_t)e * D_);
  float acc = 0.f;
  for (int k = 0; k < D_ / 4; ++k) {
    v4f a = q4[k], b = w4[k];
    v4f p = a * b;
    acc += p.x + p.y + p.z + p.w;
  }
  Qrow[e] = (acc + bq[e]) * 0.0625f;  // 1/sqrt(256)
  __syncthreads();
  const int d = threadIdx.x;
  float s = 0.f;
  for (int ee = 0; ee < D_; ++ee) s += Qrow[ee] * Wk[(size_t)ee * D_ + d];
  qk[(size_t)m * D_ + d] = s;
}

// ---------------------------------------------------------------------------
// Kernel 2: Wfv[f,d] = sum_e Wf[f,e]*Wv[e,d];  bfv[f] = Wf[f,:]·bv + bf[f]
// 256 blocks (f), 256 threads (d).
// ---------------------------------------------------------------------------
__global__ __launch_bounds__(256) void wfv_kernel(
    const float* __restrict__ Wf, const float* __restrict__ Wv,
    const float* __restrict__ bv, const float* __restrict__ bf,
    float* __restrict__ Wfv, float* __restrict__ bfv) {
  const int f = blockIdx.x, d = threadIdx.x;
  float s = 0.f;
  for (int e = 0; e < D_; ++e) s += Wf[(size_t)f * D_ + e] * Wv[(size_t)e * D_ + d];
  Wfv[(size_t)f * D_ + d] = s;
  if (d == 0) {
    float sb = bf[f];
    for (int e = 0; e < D_; ++e) sb += Wf[(size_t)f * D_ + e] * bv[e];
    bfv[f] = sb;
  }
}

// ---------------------------------------------------------------------------
// Kernel 3: the memory-bound streaming pass. One block per (b,i,j).
//   s[t]   = qk[b,i,:] · key[b,i,j,t,:]
//   attn   = softmax_t(s)
//   cv[d]  = sum_t attn[t] * value[b,i,j,t,d]
// Reads 64 KB/block (key+value) non-temporally: 512 MB total ⇒ ~22 µs at HBM.
// ---------------------------------------------------------------------------
__global__ __launch_bounds__(256) void attn_stream_kernel(
    const float* __restrict__ key, const float* __restrict__ value,
    const float* __restrict__ qk, float* __restrict__ cv) {
  __shared__ float s_attn[T_];
  __shared__ v4f s_part[4][64];

  const int m   = blockIdx.x;        // (b*L+i)*L + j
  const int bi  = m / L_;            // b*L + i
  const int tid = threadIdx.x;
  const int w   = tid >> 5;          // wave 0..7
  const int l   = tid & 31;          // lane

  const v4f* qk4 = (const v4f*)(qk + (size_t)bi * D_);
  const v4f  qa  = qk4[l];
  const v4f  qb  = qk4[l + 32];
  const v4f* k4  = (const v4f*)(key + (size_t)m * T_ * D_);
  const v4f* v4p = (const v4f*)(value + (size_t)m * T_ * D_);

  // --- scores: wave w computes t = 4w..4w+3; lane covers 2 of 64 float4 cols
  for (int q = 0; q < 4; ++q) {
    const int t = w * 4 + q;
    __builtin_prefetch(&v4p[t * (D_ / 4) + l], 0, 0);  // warm value stream
    v4f ka = __builtin_nontemporal_load(&k4[t * (D_ / 4) + l]);
    v4f kb = __builtin_nontemporal_load(&k4[t * (D_ / 4) + 32 + l]);
    v4f pr = ka * qa + kb * qb;
    float p = pr.x + pr.y + pr.z + pr.w;
    for (int off = 16; off > 0; off >>= 1) p += __shfl_xor(p, off, 32);
    if (l == 0) s_attn[t] = p;
  }
  __syncthreads();

  // --- softmax over T=32 in wave 0
  if (tid < 32) {
    float s = s_attn[tid];
    float mx = s;
    for (int off = 16; off > 0; off >>= 1) mx = fmaxf(mx, __shfl_xor(mx, off, 32));
    float e = __expf(s - mx);
    float sum = e;
    for (int off = 16; off > 0; off >>= 1) sum += __shfl_xor(sum, off, 32);
    s_attn[tid] = e / sum;
  }
  __syncthreads();

  // --- cv: 4 t-groups × 64 float4 columns
  const int g = tid >> 6, c = tid & 63;
  v4f acc = {};
  for (int tt = 0; tt < 8; ++tt) {
    const int t = g * 8 + tt;
    v4f vv = __builtin_nontemporal_load(&v4p[t * (D_ / 4) + c]);
    acc += vv * s_attn[t];
  }
  s_part[g][c] = acc;
  __syncthreads();
  if (tid < 64) {
    v4f r = s_part[0][tid] + s_part[1][tid] + s_part[2][tid] + s_part[3][tid];
    ((v4f*)(cv + (size_t)m * D_))[tid] = r;
  }
}

// ---------------------------------------------------------------------------
// Kernel 4: out[m,f] = sum_d cv[m,d]*Wfv[f,d] + bfv[f]  via V_WMMA_F32_16X16X4_F32
// M=8192, N=256, K=256. Block = 16 waves; wave w owns N-tile w; block owns a
// 16-row M tile. ISA §7.12.2 layouts: A lane=M, B lane=N, VGPR=K pair,
// half-wave selects K {0,1} vs {2,3}; D VGPR r ⇒ M=r / M=r+8 per half-wave.
// ---------------------------------------------------------------------------
__global__ __launch_bounds__(512) void out_gemm_wmma(
    const float* __restrict__ cv, const float* __restrict__ Wfv,
    const float* __restrict__ bfv, float* __restrict__ out) {
  const int m0   = blockIdx.x * 16;
  const int wave = threadIdx.x >> 5;   // 0..15 -> N tile
  const int lane = threadIdx.x & 31;
  const int n0   = wave * 16;
  const int half = lane >> 4;          // 0: K 0..1, 1: K 2..3
  const int l15  = lane & 15;

  const float* arow = cv  + (size_t)(m0 + l15) * D_ + 2 * half;
  const float* brow = Wfv + (size_t)(n0 + l15) * D_ + 2 * half;

  v8f c = {};
  for (int k0 = 0; k0 < D_; k0 += 4) {
    v2f a = *(const v2f*)(arow + k0);
    v2f b = *(const v2f*)(brow + k0);
    c = __builtin_amdgcn_wmma_f32_16x16x4_f32(false, a, false, b,
                                              (short)0, c, false, false);
  }

  const int col = n0 + l15;
  const float bias = bfv[col];
  for (int r = 0; r < 8; ++r) {
    const int row = m0 + r + 8 * half;
    out[(size_t)row * D_ + col] = c[r] + bias;
  }
}

// ---------------------------------------------------------------------------
extern "C" void kernel_launch(void* const* d_in, const int* in_sizes, int n_in,
                              void* d_out, int out_size, void* d_ws, size_t ws_size,
                              hipStream_t stream) {
  const float* key   = (const float*)d_in[0];
  const float* value = (const float*)d_in[1];
  const float* query = (const float*)d_in[2];
  const float* Wk    = (const float*)d_in[3];
  // d_in[4] = bk: softmax shift-invariant -> unused
  const float* Wv    = (const float*)d_in[5];
  const float* bv    = (const float*)d_in[6];
  const float* Wq    = (const float*)d_in[7];
  const float* bq    = (const float*)d_in[8];
  const float* Wf    = (const float*)d_in[9];
  const float* bf    = (const float*)d_in[10];
  float* out = (float*)d_out;

  float* ws  = (float*)d_ws;
  float* qk  = ws;                 // 128*256
  float* Wfv = qk + 128 * D_;      // 256*256
  float* bfv = Wfv + D_ * D_;      // 256
  float* cv  = bfv + D_;           // 8192*256  (total ~8.9 MB of ws)

  qk_fuse_kernel<<<B_ * L_, 256, 0, stream>>>(query, Wq, bq, Wk, qk);
  wfv_kernel<<<D_, 256, 0, stream>>>(Wf, Wv, bv, bf, Wfv, bfv);
  attn_stream_kernel<<<B_ * L_ * L_, 256, 0, stream>>>(key, value, qk, cv);
  out_gemm_wmma<<<(B_ * L_ * L_) / 16, 512, 0, stream>>>(cv, Wfv, bfv, out);
}